// GAT_dialoggcn_v1_67087389164134
// MI455X (gfx1250) — compile-verified
//
#include <hip/hip_runtime.h>

#define BSZ 32
#define NN 2048
#define DD 512
#define KDD 256
#define XD 768            // DD + KDD
#define CSTRIDE 1538      // 2*XD + 2 (c0 | c1 | s0 | s1)
#define NCHUNK 8

typedef float v2f __attribute__((ext_vector_type(2)));
typedef float v8f __attribute__((ext_vector_type(8)));

#if __has_builtin(__builtin_amdgcn_wmma_f32_16x16x4_f32)
#define HAVE_WMMA_F32 1
#else
#define HAVE_WMMA_F32 0
#endif

__device__ inline float wave_reduce_sum(float v) {
  #pragma unroll
  for (int off = 16; off > 0; off >>= 1) v += __shfl_xor(v, off, 32);
  return v;
}

__device__ inline void fma4(float4& acc, float c, const float4& x) {
  acc.x += c * x.x; acc.y += c * x.y; acc.z += c * x.z; acc.w += c * x.w;
}

// ---------------------------------------------------------------------------
// Kernel 1: v = W @ wden[D:]  (768-vector per branch),
//           base[b] = Q_b . wden[:D] + bias . wden[D:] + bden
// One wave per output row. 1600 waves total.
// ---------------------------------------------------------------------------
__global__ void prep_kernel(const float* __restrict__ Q,
                            const float* __restrict__ Wfk, const float* __restrict__ bfk,
                            const float* __restrict__ Wbk, const float* __restrict__ bbk,
                            const float* __restrict__ wf, const float* __restrict__ bf,
                            const float* __restrict__ wb, const float* __restrict__ bb,
                            float* __restrict__ vf, float* __restrict__ vb,
                            float* __restrict__ basef, float* __restrict__ baseb) {
  int wid  = (blockIdx.x * blockDim.x + threadIdx.x) >> 5;
  int lane = threadIdx.x & 31;
  if (wid < 2 * XD) {
    int br = wid >= XD;
    int j  = br ? wid - XD : wid;
    const float4* W4 = (const float4*)((br ? Wbk : Wfk) + (size_t)j * DD);
    const float4* w4 = (const float4*)((br ? wb : wf) + DD);
    float acc = 0.f;
    #pragma unroll
    for (int q = 0; q < 4; ++q) {
      float4 a = W4[lane + 32 * q], c = w4[lane + 32 * q];
      acc += a.x * c.x + a.y * c.y + a.z * c.z + a.w * c.w;
    }
    acc = wave_reduce_sum(acc);
    if (lane == 0) (br ? vb : vf)[j] = acc;
  } else {
    int t  = wid - 2 * XD;          // 0..63
    int br = t >= BSZ;
    int b_ = br ? t - BSZ : t;
    const float*  wd  = br ? wb : wf;
    const float4* Q4  = (const float4*)(Q + (size_t)b_ * DD);
    const float4* w1  = (const float4*)wd;
    const float4* w2  = (const float4*)(wd + DD);
    const float4* bi4 = (const float4*)(br ? bbk : bfk);
    float acc = 0.f;
    #pragma unroll
    for (int q = 0; q < 4; ++q) {
      float4 a = Q4[lane + 32 * q], c = w1[lane + 32 * q];
      acc += a.x * c.x + a.y * c.y + a.z * c.z + a.w * c.w;
      float4 d = bi4[lane + 32 * q], e = w2[lane + 32 * q];
      acc += d.x * e.x + d.y * e.y + d.z * e.z + d.w * e.w;
    }
    acc = wave_reduce_sum(acc);
    if (lane == 0) (br ? baseb : basef)[b_] = acc + (br ? bb[0] : bf[0]);
  }
}

// ---------------------------------------------------------------------------
// Kernel 2: alpha[b,n] = x[b,n] . v + base[b] - (1-adj)*1e30
// One wave per row (65536 rows). Streaming, coalesced float4 loads.
// ---------------------------------------------------------------------------
__global__ void alpha_kernel(const float* __restrict__ Kv, const float* __restrict__ kf,
                             const float* __restrict__ v, const float* __restrict__ base,
                             const int* __restrict__ adj, float* __restrict__ alpha) {
  int wid  = (blockIdx.x * blockDim.x + threadIdx.x) >> 5;
  int lane = threadIdx.x & 31;
  int b    = wid >> 11;
  const float4* K4 = (const float4*)(Kv + (size_t)wid * DD);
  const float4* k4 = (const float4*)(kf + (size_t)wid * KDD);
  const float4* vK = (const float4*)v;
  const float4* vk = (const float4*)(v + DD);
  float acc = 0.f;
  #pragma unroll
  for (int q = 0; q < 4; ++q) {
    float4 a = K4[lane + 32 * q], c = vK[lane + 32 * q];
    acc += a.x * c.x + a.y * c.y + a.z * c.z + a.w * c.w;
  }
  #pragma unroll
  for (int q = 0; q < 2; ++q) {
    float4 a = k4[lane + 32 * q], c = vk[lane + 32 * q];
    acc += a.x * c.x + a.y * c.y + a.z * c.z + a.w * c.w;
  }
  acc = wave_reduce_sum(acc);
  if (lane == 0) {
    float r = acc + base[b];
    if (adj[wid] == 0) r -= 1e30f;
    alpha[wid] = r;
  }
}

// ---------------------------------------------------------------------------
// Kernel 3: in-place masked softmax over N per batch row (32 blocks).
// ---------------------------------------------------------------------------
__global__ void softmax_kernel(float* __restrict__ alpha) {
  __shared__ float red[256];
  float* a = alpha + (size_t)blockIdx.x * NN;
  int tid = threadIdx.x;
  float vals[8];
  float mx = -3.4e38f;
  #pragma unroll
  for (int j = 0; j < 8; ++j) { vals[j] = a[tid + 256 * j]; mx = fmaxf(mx, vals[j]); }
  red[tid] = mx; __syncthreads();
  for (int s = 128; s > 0; s >>= 1) { if (tid < s) red[tid] = fmaxf(red[tid], red[tid + s]); __syncthreads(); }
  mx = red[0]; __syncthreads();
  float sum = 0.f;
  #pragma unroll
  for (int j = 0; j < 8; ++j) { vals[j] = __expf(vals[j] - mx); sum += vals[j]; }
  red[tid] = sum; __syncthreads();
  for (int s = 128; s > 0; s >>= 1) { if (tid < s) red[tid] += red[tid + s]; __syncthreads(); }
  float inv = 1.f / red[0];
  #pragma unroll
  for (int j = 0; j < 8; ++j) a[tid + 256 * j] = vals[j] * inv;
}

// ---------------------------------------------------------------------------
// Kernel 4: c0[b] = sum_n w*sm*x[b,n], c1[b] = sum_n w*(1-sm)*x[b,n],
//           s0 = sum w*sm, s1 = sum w*(1-sm).
// Block = (b, chunk of 256 rows); 8 waves/block, one row per wave per step,
// register accumulators per lane, deterministic LDS tree reduction.
// ---------------------------------------------------------------------------
__global__ void wsum_kernel(const float* __restrict__ Kv, const float* __restrict__ kf,
                            const float* __restrict__ w, const int* __restrict__ smask,
                            float* __restrict__ partials) {
  __shared__ float lacc[8][CSTRIDE];
  int b     = blockIdx.x >> 3;
  int chunk = blockIdx.x & 7;
  int wave  = threadIdx.x >> 5;
  int lane  = threadIdx.x & 31;
  float4 a0K[4] = {}; float4 a1K[4] = {};
  float4 a0k[2] = {}; float4 a1k[2] = {};
  float s0 = 0.f, s1 = 0.f;
  int n0 = chunk * 256 + wave * 32;
  for (int t = 0; t < 32; ++t) {
    size_t row = (size_t)b * NN + (n0 + t);
    float wgt = w[row];
    int   sm  = smask[row];
    float c0  = sm ? wgt : 0.f;
    float c1  = wgt - c0;
    s0 += c0; s1 += c1;
    const float4* K4 = (const float4*)(Kv + row * DD);
    const float4* k4 = (const float4*)(kf + row * KDD);
    #pragma unroll
    for (int q = 0; q < 4; ++q) {
      float4 x = K4[lane + 32 * q];
      fma4(a0K[q], c0, x); fma4(a1K[q], c1, x);
    }
    #pragma unroll
    for (int q = 0; q < 2; ++q) {
      float4 x = k4[lane + 32 * q];
      fma4(a0k[q], c0, x); fma4(a1k[q], c1, x);
    }
  }
  #pragma unroll
  for (int q = 0; q < 4; ++q) {
    int base = (lane + 32 * q) * 4;
    lacc[wave][base + 0] = a0K[q].x; lacc[wave][base + 1] = a0K[q].y;
    lacc[wave][base + 2] = a0K[q].z; lacc[wave][base + 3] = a0K[q].w;
    lacc[wave][XD + base + 0] = a1K[q].x; lacc[wave][XD + base + 1] = a1K[q].y;
    lacc[wave][XD + base + 2] = a1K[q].z; lacc[wave][XD + base + 3] = a1K[q].w;
  }
  #pragma unroll
  for (int q = 0; q < 2; ++q) {
    int base = DD + (lane + 32 * q) * 4;
    lacc[wave][base + 0] = a0k[q].x; lacc[wave][base + 1] = a0k[q].y;
    lacc[wave][base + 2] = a0k[q].z; lacc[wave][base + 3] = a0k[q].w;
    lacc[wave][XD + base + 0] = a1k[q].x; lacc[wave][XD + base + 1] = a1k[q].y;
    lacc[wave][XD + base + 2] = a1k[q].z; lacc[wave][XD + base + 3] = a1k[q].w;
  }
  if (lane == 0) { lacc[wave][1536] = s0; lacc[wave][1537] = s1; }
  __syncthreads();
  float* outp = partials + ((size_t)b * NCHUNK + chunk) * CSTRIDE;
  for (int idx = threadIdx.x; idx < CSTRIDE; idx += 256) {
    float s = 0.f;
    #pragma unroll
    for (int wv = 0; wv < 8; ++wv) s += lacc[wv][idx];
    outp[idx] = s;
  }
}

// ---------------------------------------------------------------------------
// Kernel 5: reduce chunk partials (fixed order => deterministic).
// ---------------------------------------------------------------------------
__global__ void reduce_partials_kernel(const float* __restrict__ partials,
                                       float* __restrict__ cout) {
  int idx = blockIdx.x * blockDim.x + threadIdx.x;
  if (idx >= BSZ * CSTRIDE) return;
  int b = idx / CSTRIDE;
  int j = idx - b * CSTRIDE;
  float s = 0.f;
  #pragma unroll
  for (int c = 0; c < NCHUNK; ++c)
    s += partials[((size_t)b * NCHUNK + c) * CSTRIDE + j];
  cout[idx] = s;
}

// ---------------------------------------------------------------------------
// Kernel 6 (WMMA): u = c @ W + s*bias, four (32x768)@(768x512) products.
// One wave per 16x16 output tile; 256 waves total. EXEC stays all-1s.
// ---------------------------------------------------------------------------
__global__ void proj_u_kernel(const float* __restrict__ cF, const float* __restrict__ cB,
                              const float* __restrict__ Wfk, const float* __restrict__ bfk,
                              const float* __restrict__ Wbk, const float* __restrict__ bbk,
                              float* __restrict__ ubase) {
  int waveG = (blockIdx.x * blockDim.x + threadIdx.x) >> 5;
  int lane  = threadIdx.x & 31;
  int half  = lane >> 4;
  int ln    = lane & 15;
  int mat   = waveG >> 6;                 // 0..3 : {f,c0} {f,c1} {b,c0} {b,c1}
  int t     = waveG & 63;
  int m0    = (t >> 5) * 16;
  int n0    = (t & 31) * 16;
  int br = mat >> 1, which = mat & 1;
  const float* C    = br ? cB : cF;
  const float* W    = br ? Wbk : Wfk;
  const float* bias = br ? bbk : bfk;
  const float* crow = C + which * XD;
  float* uout = ubase + (size_t)mat * (BSZ * DD);

  v8f acc = {0.f, 0.f, 0.f, 0.f, 0.f, 0.f, 0.f, 0.f};
#if HAVE_WMMA_F32
  for (int k = 0; k < XD; k += 4) {
    v2f a, bb_;
    const float* ar = crow + (size_t)(m0 + ln) * CSTRIDE + k + 2 * half;
    a.x = ar[0];
    a.y = ar[1];
    bb_.x = W[(size_t)(k + 2 * half)     * DD + n0 + ln];
    bb_.y = W[(size_t)(k + 2 * half + 1) * DD + n0 + ln];
    acc = __builtin_amdgcn_wmma_f32_16x16x4_f32(false, a, false, bb_,
                                                (short)0, acc, false, false);
  }
#else
  for (int r = 0; r < 8; ++r) {
    int m = m0 + r + 8 * half, n = n0 + ln;
    float s = 0.f;
    for (int k = 0; k < XD; ++k) s += crow[(size_t)m * CSTRIDE + k] * W[(size_t)k * DD + n];
    acc[r] = s;
  }
#endif
  #pragma unroll
  for (int r = 0; r < 8; ++r) {
    int m = m0 + r + 8 * half, n = n0 + ln;
    float sv = C[(size_t)m * CSTRIDE + 1536 + which];
    uout[(size_t)m * DD + n] = acc[r] + sv * bias[n];
  }
}

// ---------------------------------------------------------------------------
// Kernel 7 (WMMA): out = u0 @ Wr0 + u1 @ Wr1 per branch, plus the i/num_utter
// zeroing condition (device-side scalars). 128 waves.
// ---------------------------------------------------------------------------
__global__ void out_kernel(const float* __restrict__ ubase,
                           const float* __restrict__ Wr0, const float* __restrict__ Wr1,
                           const int* __restrict__ ip, const int* __restrict__ nup,
                           float* __restrict__ out) {
  int waveG = (blockIdx.x * blockDim.x + threadIdx.x) >> 5;
  int lane  = threadIdx.x & 31;
  int half  = lane >> 4, ln = lane & 15;
  int br    = waveG >> 6;
  int t     = waveG & 63;
  int m0    = (t >> 5) * 16, n0 = (t & 31) * 16;
  const float* u0 = ubase + (size_t)br * 2 * (BSZ * DD);
  const float* u1 = u0 + BSZ * DD;

  v8f acc = {0.f, 0.f, 0.f, 0.f, 0.f, 0.f, 0.f, 0.f};
#if HAVE_WMMA_F32
  for (int k = 0; k < DD; k += 4) {
    v2f a, bb_;
    const float* ar = u0 + (size_t)(m0 + ln) * DD + k + 2 * half;
    a.x = ar[0]; a.y = ar[1];
    bb_.x = Wr0[(size_t)(k + 2 * half)     * DD + n0 + ln];
    bb_.y = Wr0[(size_t)(k + 2 * half + 1) * DD + n0 + ln];
    acc = __builtin_amdgcn_wmma_f32_16x16x4_f32(false, a, false, bb_,
                                                (short)0, acc, false, false);
  }
  for (int k = 0; k < DD; k += 4) {
    v2f a, bb_;
    const float* ar = u1 + (size_t)(m0 + ln) * DD + k + 2 * half;
    a.x = ar[0]; a.y = ar[1];
    bb_.x = Wr1[(size_t)(k + 2 * half)     * DD + n0 + ln];
    bb_.y = Wr1[(size_t)(k + 2 * half + 1) * DD + n0 + ln];
    acc = __builtin_amdgcn_wmma_f32_16x16x4_f32(false, a, false, bb_,
                                                (short)0, acc, false, false);
  }
#else
  for (int r = 0; r < 8; ++r) {
    int m = m0 + r + 8 * half, n = n0 + ln;
    float s = 0.f;
    for (int k = 0; k < DD; ++k)
      s += u0[(size_t)m * DD + k] * Wr0[(size_t)k * DD + n]
         + u1[(size_t)m * DD + k] * Wr1[(size_t)k * DD + n];
    acc[r] = s;
  }
#endif
  int iv = ip[0], nu = nup[0];
  bool zero = (br == 0) ? (iv == 0) : (iv == nu - 1);
  #pragma unroll
  for (int r = 0; r < 8; ++r) {
    int m = m0 + r + 8 * half, n = n0 + ln;
    out[(size_t)br * (BSZ * DD) + (size_t)m * DD + n] = zero ? 0.f : acc[r];
  }
}

// ---------------------------------------------------------------------------
extern "C" void kernel_launch(void* const* d_in, const int* in_sizes, int n_in,
                              void* d_out, int out_size, void* d_ws, size_t ws_size,
                              hipStream_t stream) {
  (void)in_sizes; (void)n_in; (void)out_size; (void)ws_size;
  const float* Q    = (const float*)d_in[0];
  const float* K    = (const float*)d_in[1];
  const float* fk1  = (const float*)d_in[3];
  const float* bK   = (const float*)d_in[7];
  const float* bk2  = (const float*)d_in[10];
  const float* Wfk  = (const float*)d_in[13];
  const float* bfk  = (const float*)d_in[14];
  const float* Wbk  = (const float*)d_in[15];
  const float* bbk  = (const float*)d_in[16];
  const float* Wr0  = (const float*)d_in[17];
  const float* Wr1  = (const float*)d_in[18];
  const float* wf   = (const float*)d_in[19];
  const float* bf   = (const float*)d_in[20];
  const float* wb   = (const float*)d_in[21];
  const float* bb   = (const float*)d_in[22];
  const int*   fadj = (const int*)d_in[23];
  const int*   fsm  = (const int*)d_in[24];
  const int*   badj = (const int*)d_in[25];
  const int*   bsm  = (const int*)d_in[26];
  const int*   ip   = (const int*)d_in[27];
  const int*   nup  = (const int*)d_in[28];

  float* ws     = (float*)d_ws;
  float* vf     = ws;                                   // 768
  float* vb     = vf + XD;                              // 768
  float* basef  = vb + XD;                              // 32
  float* baseb  = basef + BSZ;                          // 32
  float* alphaf = baseb + BSZ;                          // 65536
  float* alphab = alphaf + BSZ * NN;                    // 65536
  float* partf  = alphab + BSZ * NN;                    // 32*8*1538
  float* partb  = partf + BSZ * NCHUNK * CSTRIDE;       // 32*8*1538
  float* cF     = partb + BSZ * NCHUNK * CSTRIDE;       // 32*1538
  float* cB     = cF + BSZ * CSTRIDE;                   // 32*1538
  float* ub     = cB + BSZ * CSTRIDE;                   // 4*32*512

  prep_kernel<<<400, 128, 0, stream>>>(Q, Wfk, bfk, Wbk, bbk, wf, bf, wb, bb,
                                       vf, vb, basef, baseb);
  alpha_kernel<<<8192, 256, 0, stream>>>(K,  fk1, vf, basef, fadj, alphaf);
  alpha_kernel<<<8192, 256, 0, stream>>>(bK, bk2, vb, baseb, badj, alphab);
  softmax_kernel<<<BSZ, 256, 0, stream>>>(alphaf);
  softmax_kernel<<<BSZ, 256, 0, stream>>>(alphab);
  wsum_kernel<<<BSZ * NCHUNK, 256, 0, stream>>>(K,  fk1, alphaf, fsm, partf);
  wsum_kernel<<<BSZ * NCHUNK, 256, 0, stream>>>(bK, bk2, alphab, bsm, partb);
  reduce_partials_kernel<<<(BSZ * CSTRIDE + 255) / 256, 256, 0, stream>>>(partf, cF);
  reduce_partials_kernel<<<(BSZ * CSTRIDE + 255) / 256, 256, 0, stream>>>(partb, cB);
  proj_u_kernel<<<32, 256, 0, stream>>>(cF, cB, Wfk, bfk, Wbk, bbk, ub);
  out_kernel<<<16, 256, 0, stream>>>(ub, Wr0, Wr1, ip, nup, (float*)d_out);
}